// Group_3006477107875
// MI455X (gfx1250) — compile-verified
//
#include <hip/hip_runtime.h>

typedef __attribute__((ext_vector_type(2))) float v2f;
typedef __attribute__((ext_vector_type(8))) float v8f;
typedef unsigned int u32x4 __attribute__((ext_vector_type(4)));
typedef int i32x4 __attribute__((ext_vector_type(4)));
typedef int i32x8 __attribute__((ext_vector_type(8)));

#define B_  32
#define N_  8192
#define G_  512
#define M_  32

#if defined(__gfx1250__) && __has_builtin(__builtin_amdgcn_tensor_load_to_lds) && \
    __has_builtin(__builtin_amdgcn_s_wait_tensorcnt)
#define HAVE_TDM 1
#else
#define HAVE_TDM 0
#endif

// ---------------------------------------------------------------------------
// Kernel 1: Furthest Point Sampling. One block per batch, 512 threads.
// Each thread owns 16 points in registers (strided n = j*512 + t).
// centers[g] = xyz[far_g]; far_0 = 0; far_{g+1} = argmax of running min-dist,
// ties -> lowest index (matches jnp.argmax).
// ---------------------------------------------------------------------------
__global__ __launch_bounds__(512) void fps_kernel(const float* __restrict__ xyz,
                                                  float* __restrict__ centers)
{
    const int b = blockIdx.x;
    const int t = threadIdx.x;
    const float* base = xyz + (size_t)b * N_ * 3;

    float px[16], py[16], pz[16], dist[16];
#pragma unroll
    for (int j = 0; j < 16; ++j) {
        int n = j * 512 + t;
        px[j] = base[n * 3 + 0];
        py[j] = base[n * 3 + 1];
        pz[j] = base[n * 3 + 2];
        dist[j] = __builtin_inff();
    }

    __shared__ float s_rv[16];
    __shared__ int   s_ri[16];
    __shared__ int   s_far;

    const int lane = t & 31;
    const int wave = t >> 5;
    int far = 0;

    for (int g = 0; g < G_; ++g) {
        float cx = base[far * 3 + 0];
        float cy = base[far * 3 + 1];
        float cz = base[far * 3 + 2];
        if (t == 0) {
            float* c = centers + ((size_t)b * G_ + g) * 3;
            c[0] = cx; c[1] = cy; c[2] = cz;
        }

        float bv = -1.0f; int bi = 0x7fffffff;
#pragma unroll
        for (int j = 0; j < 16; ++j) {
            float dx = px[j] - cx, dy = py[j] - cy, dz = pz[j] - cz;
            float d  = dx * dx + dy * dy + dz * dz;
            float nd = dist[j] < d ? dist[j] : d;
            dist[j] = nd;
            int n = j * 512 + t;
            if (nd > bv || (nd == bv && n < bi)) { bv = nd; bi = n; }
        }
        for (int off = 16; off > 0; off >>= 1) {
            float ov = __shfl_down(bv, (unsigned)off);
            int   oi = __shfl_down(bi, (unsigned)off);
            if (ov > bv || (ov == bv && oi < bi)) { bv = ov; bi = oi; }
        }
        if (lane == 0) { s_rv[wave] = bv; s_ri[wave] = bi; }
        __syncthreads();
        if (t == 0) {
            float mv = s_rv[0]; int mi = s_ri[0];
            for (int w = 1; w < 16; ++w) {
                if (s_rv[w] > mv || (s_rv[w] == mv && s_ri[w] < mi)) {
                    mv = s_rv[w]; mi = s_ri[w];
                }
            }
            s_far = mi;
        }
        __syncthreads();
        far = s_far;
        __syncthreads();
    }
}

// ---------------------------------------------------------------------------
// Kernel 2: kNN (top-32 smallest sq-dist) + gather + center-subtract.
// Grid: (G/16, B). Block: 128 threads = 4 waves.
// Pipeline per 64-point step:
//   TDM (tensor_load_to_lds) prefetches the NEXT 768B point chunk into the
//   shadow LDS buffer while WMMA consumes the current one:
//     V_WMMA_F32_16X16X4_F32: A[m][:]=[-2cx,-2cy,-2cz,1], B[:][n]=[x,y,z,|x|^2]
//     => acc = |x|^2 - 2 c.x ; d = acc + |c|^2
//   wave0 issues TDM, waits s_wait_tensorcnt 0 before the step-end barrier.
// ---------------------------------------------------------------------------
__global__ __launch_bounds__(128) void knn_kernel(const float* __restrict__ xyz,
                                                  const float* __restrict__ centers,
                                                  float* __restrict__ neigh)
{
    const int tile = blockIdx.x;
    const int b    = blockIdx.y;
    const int t    = threadIdx.x;          // 0..127
    const int lane = t & 31;
    const int wave = t >> 5;               // 0..3
    const int half = lane >> 4;
    const int l16  = lane & 15;

    const float* base = xyz + (size_t)b * N_ * 3;
    const float* ctr  = centers + ((size_t)b * G_ + (size_t)tile * 16) * 3;

    __shared__ float s_pts[2][200];        // double-buffered 64-pt chunks (192 used)
    __shared__ float s_cx[16], s_cy[16], s_cz[16], s_cn[16];
    __shared__ float s_d[16][65];
    __shared__ float s_ld[16][M_];
    __shared__ int   s_li[16][M_];
    __shared__ float s_worst[16];
    __shared__ int   s_wpos[16];

    // Issue a TDM DMA (or fallback copy) of chunk -> s_pts[bufid]
    auto load_chunk = [&](int bufid, int chunk) {
#if HAVE_TDM
        if (t == 0) {
            uint64_t ga  = (uint64_t)(uintptr_t)(base + (size_t)chunk * 192);
            uint32_t lds = (uint32_t)(uintptr_t)&s_pts[bufid][0]; // low32 == LDS offset
            u32x4 g0;
            g0[0] = 1u;                                   // count=1 (valid D#)
            g0[1] = lds;                                  // lds_addr
            g0[2] = (uint32_t)(ga & 0xffffffffu);         // global_addr[31:0]
            g0[3] = (uint32_t)((ga >> 32) & 0x01ffffffu)  // global_addr[56:32]
                    | (2u << 30);                         // type=2 (image)
            i32x8 g1;
            g1[0] = (int)(2u << 16);                      // data_size=4B
            g1[1] = (int)((24576u & 0xffffu) << 16);      // tensor_dim0[15:0]
            g1[2] = (int)((24576u >> 16) | (1u << 16));   // td0[31:16] | tensor_dim1=1
            g1[3] = (int)(192u << 16);                    // tile_dim0=192
            g1[4] = 1;                                    // tile_dim1=1, tile_dim2=0
            g1[5] = 24576;                                // tensor_dim0_stride lo32
            g1[6] = 0;                                    // stride0 hi / stride1 lo
            g1[7] = 0;
            i32x4 gz = {0, 0, 0, 0};
#if __clang_major__ >= 23
            i32x8 gz8 = {0, 0, 0, 0, 0, 0, 0, 0};
            __builtin_amdgcn_tensor_load_to_lds(g0, g1, gz, gz, gz8, 0);
#else
            __builtin_amdgcn_tensor_load_to_lds(g0, g1, gz, gz, 0);
#endif
        }
#else
        for (int e = t; e < 192; e += 128)
            s_pts[bufid][e] = base[(size_t)chunk * 192 + e];
#endif
    };
    auto wait_chunk = [&]() {
#if HAVE_TDM
        if (t == 0) __builtin_amdgcn_s_wait_tensorcnt(0);
#endif
    };

    if (t < 16) {
        float x = ctr[t * 3 + 0], y = ctr[t * 3 + 1], z = ctr[t * 3 + 2];
        s_cx[t] = x; s_cy[t] = y; s_cz[t] = z;
        s_cn[t] = x * x + y * y + z * z;
        for (int k = 0; k < M_; ++k) { s_ld[t][k] = __builtin_inff(); s_li[t][k] = 0; }
        s_worst[t] = __builtin_inff();
        s_wpos[t]  = 0;
    }
    load_chunk(0, 0);
    wait_chunk();
    __syncthreads();

    // A operand (identical across waves): lanes 0-15 hold K=0,1; 16-31 hold K=2,3
    v2f A;
    if (half == 0) { A[0] = -2.0f * s_cx[l16]; A[1] = -2.0f * s_cy[l16]; }
    else           { A[0] = -2.0f * s_cz[l16]; A[1] = 1.0f; }
    float cn[8];
#pragma unroll
    for (int r = 0; r < 8; ++r) cn[r] = s_cn[r + 8 * half];

    const int NSTEP = N_ / 64;                        // 128 steps of 64 points
    for (int it = 0; it < NSTEP; ++it) {
        const int cur = it & 1;
        if (it + 1 < NSTEP) load_chunk(cur ^ 1, it + 1);

        const int p = wave * 16 + l16;                // point within chunk
        float qx = s_pts[cur][p * 3 + 0];
        float qy = s_pts[cur][p * 3 + 1];
        float qz = s_pts[cur][p * 3 + 2];
        v2f Bm;
        if (half == 0) { Bm[0] = qx; Bm[1] = qy; }
        else           { Bm[0] = qz; Bm[1] = qx * qx + qy * qy + qz * qz; }

        v8f acc = {};
        acc = __builtin_amdgcn_wmma_f32_16x16x4_f32(
                  false, A, false, Bm, (short)0, acc, false, false);

#pragma unroll
        for (int r = 0; r < 8; ++r) {
            int m = r + 8 * half;
            s_d[m][wave * 16 + l16] = acc[r] + cn[r];
        }
        __syncthreads();

        // Threshold-filtered running top-32 per center (thread t == center m)
        if (t < 16) {
            const int baseIdx = it * 64;
            float worst = s_worst[t];
            int   wp    = s_wpos[t];
            for (int c = 0; c < 64; ++c) {
                float dd = s_d[t][c];
                if (dd < worst) {
                    s_ld[t][wp] = dd;
                    s_li[t][wp] = baseIdx + c;
                    float mv = s_ld[t][0]; int mp = 0;
                    for (int k = 1; k < M_; ++k) {
                        float v = s_ld[t][k];
                        if (v > mv) { mv = v; mp = k; }
                    }
                    worst = mv; wp = mp;
                }
            }
            s_worst[t] = worst;
            s_wpos[t]  = wp;
        }
        wait_chunk();          // next buffer resident before releasing waves
        __syncthreads();
    }

    // Sort each center's 32 hits ascending by (dist, idx) to match lax.top_k
    if (t < 16) {
        for (int i = 0; i < M_ - 1; ++i) {
            int best = i;
            for (int k = i + 1; k < M_; ++k) {
                float vk = s_ld[t][k], vb = s_ld[t][best];
                if (vk < vb || (vk == vb && s_li[t][k] < s_li[t][best])) best = k;
            }
            if (best != i) {
                float tv = s_ld[t][i]; s_ld[t][i] = s_ld[t][best]; s_ld[t][best] = tv;
                int   ti = s_li[t][i]; s_li[t][i] = s_li[t][best]; s_li[t][best] = ti;
            }
        }
    }
    __syncthreads();

    // Gather neighbors, subtract center, write neighborhood [B,G,M,3]
    for (int e = t; e < 16 * M_; e += 128) {
        int m = e / M_, k = e % M_;
        int idx = s_li[m][k];
        float ox = base[idx * 3 + 0] - s_cx[m];
        float oy = base[idx * 3 + 1] - s_cy[m];
        float oz = base[idx * 3 + 2] - s_cz[m];
        size_t g = (size_t)tile * 16 + m;
        float* o = neigh + (((size_t)b * G_ + g) * (size_t)M_ + k) * 3;
        o[0] = ox; o[1] = oy; o[2] = oz;
    }
}

// ---------------------------------------------------------------------------
extern "C" void kernel_launch(void* const* d_in, const int* in_sizes, int n_in,
                              void* d_out, int out_size, void* d_ws, size_t ws_size,
                              hipStream_t stream) {
    (void)in_sizes; (void)n_in; (void)out_size; (void)d_ws; (void)ws_size;
    const float* xyz = (const float*)d_in[0];
    // d_out layout: neighborhood [B,G,M,3] then centers [B,G,3]
    float* neigh   = (float*)d_out;
    float* centers = (float*)d_out + (size_t)B_ * G_ * M_ * 3;

    fps_kernel<<<dim3(B_), dim3(512), 0, stream>>>(xyz, centers);
    knn_kernel<<<dim3(G_ / 16, B_), dim3(128), 0, stream>>>(xyz, centers, neigh);
}